// MambaScanner_42339787604670
// MI455X (gfx1250) — compile-verified
//
#include <hip/hip_runtime.h>
#include <math.h>

// MambaScanner for MI455X (gfx1250, wave32, WMMA + async-to-LDS).
// Shapes: B=4, K=4, D=96, L=2048, N=16, R=6, C=R+2N=38.
//
// Pipeline:
//   proj1: x_dbl = W1(CxD) * xs(DxL) via V_WMMA_F32_16X16X4_F32, 4 N-tiles/wave.
//          rows c<6 -> (c,l) staging for proj2; rows 6..37 -> (l,32) transposed for scan.
//   proj2: dt = W2(DxR) * dts(RxL) via WMMA (R padded 6->8); epilogue fuses
//          softplus and stores (delta, delta*x) pairs  -> scan's critical chain
//          is just v_exp_f32 + FMA.
//   scan:  block = 8 waves, uniform (b,k); B/C stream double-buffered into LDS
//          with global_load_async_to_lds_b128 (ASYNCcnt); lanes = (n, d-parity);
//          16-wide shfl_xor tree for sum_n h*C.

typedef __attribute__((ext_vector_type(2))) float v2f;
typedef __attribute__((ext_vector_type(8))) float v8f;

#define B_ 4
#define K_ 4
#define D_ 96
#define L_ 2048
#define N_ 16
#define R_ 6
#define C_ 38

#define CHUNK 64                       // l-steps staged per LDS buffer (8 KB)

// workspace layout (floats)
#define DTS_CL_ELEMS  (B_*K_*8*L_)     // dt rows, (c,l), padded to 8 rows
#define BC_T_ELEMS    (B_*K_*L_*32)    // B/C rows transposed, (l,32) = [B(16)|C(16)]
// dd pairs follow: B_*K_*D_*L_*2  (delta, delta*x)

// ---------------------------------------------------------------------------
// Kernel 1: x_dbl = W1 * xs, one A-fragment reused across 4 N-tiles.
// waves = B*K * 3 Mtiles * 32 Ngroups = 1536 ; 8 waves/block -> 192 blocks
// ---------------------------------------------------------------------------
__global__ __launch_bounds__(256) void proj1_kernel(const float* __restrict__ xs,
                                                    const float* __restrict__ xpw,
                                                    float* __restrict__ dts_cl,
                                                    float* __restrict__ bc_t) {
  const int wave = blockIdx.x * 8 + (threadIdx.x >> 5);
  const int lane = threadIdx.x & 31;
  const int half = lane >> 4;
  const int m    = lane & 15;

  const int ntg = wave & 31;               // group of 4 L-tiles (64 columns)
  const int mt  = (wave >> 5) % 3;         // C tile
  const int k   = (wave / (32 * 3)) & 3;
  const int b   = wave / (32 * 3 * 4);

  const int l0   = ntg * 64 + m;
  const int crow = mt * 16 + m;
  const int crc  = (crow < C_) ? crow : 0;       // clamped (branchless pad)
  const float msk = (crow < C_) ? 1.0f : 0.0f;

  const float* xsbk = xs + ((size_t)(b * K_ + k)) * D_ * L_;
  const float* w    = xpw + (size_t)k * (C_ * D_) + (size_t)crc * D_;

  v8f acc[4] = {};
  for (int kk = 0; kk < D_ / 4; ++kk) {
    const int d0 = kk * 4 + 2 * half;            // K = 2*half + j (ISA 16x4 layout)
    v2f a;
    a.x = w[d0] * msk;
    a.y = w[d0 + 1] * msk;
    const float* xp0 = xsbk + (size_t)d0 * L_ + l0;
    const float* xp1 = xp0 + L_;
    #pragma unroll
    for (int t = 0; t < 4; ++t) {
      v2f bm;
      bm.x = xp0[t * 16];
      bm.y = xp1[t * 16];
      acc[t] = __builtin_amdgcn_wmma_f32_16x16x4_f32(false, a, false, bm,
                                                     (short)0, acc[t], false, false);
    }
  }

  #pragma unroll
  for (int t = 0; t < 4; ++t) {
    const int l = l0 + t * 16;
    #pragma unroll
    for (int v = 0; v < 8; ++v) {
      const int c = mt * 16 + v + 8 * half;      // C/D: VGPR v -> row v + 8*half
      const float val = acc[t][v];
      if (c < R_) {
        dts_cl[((size_t)(b * K_ + k) * 8 + c) * L_ + l] = val;
      } else if (c < C_) {
        bc_t[((size_t)(b * K_ + k) * L_ + l) * 32 + (c - R_)] = val;
      }
    }
  }
}

// ---------------------------------------------------------------------------
// Kernel 2: dt = W2 * dts (WMMA, R padded 6->8); fused softplus epilogue.
// Stores dd[(bk*D+d)*L + l] = { delta, delta*x } as float2.
// waves = B*K * 6 Mtiles * 128 Ntiles = 12288 ; 1536 blocks
// ---------------------------------------------------------------------------
__global__ __launch_bounds__(256) void proj2_kernel(const float* __restrict__ dts_cl,
                                                    const float* __restrict__ dtw,
                                                    const float* __restrict__ dtb,
                                                    const float* __restrict__ xs,
                                                    float* __restrict__ dd) {
  const int wave = blockIdx.x * 8 + (threadIdx.x >> 5);
  const int lane = threadIdx.x & 31;
  const int half = lane >> 4;
  const int m    = lane & 15;

  const int nt = wave & 127;
  const int mt = (wave >> 7) % 6;
  const int k  = (wave / (128 * 6)) & 3;
  const int b  = wave / (128 * 6 * 4);

  const int l    = nt * 16 + m;
  const int drow = mt * 16 + m;

  const float* wrow = dtw + (size_t)(k * D_ + drow) * R_;        // (K,D,R)
  const float* dbk  = dts_cl + (size_t)(b * K_ + k) * 8 * L_;

  v8f acc = {};
  #pragma unroll
  for (int kk = 0; kk < 2; ++kk) {
    const int r0 = kk * 4 + 2 * half;
    v2f a, bm;
    a.x  = (r0     < R_) ? wrow[r0]                       : 0.0f;
    a.y  = (r0 + 1 < R_) ? wrow[r0 + 1]                   : 0.0f;
    bm.x = (r0     < R_) ? dbk[(size_t)r0 * L_ + l]       : 0.0f;
    bm.y = (r0 + 1 < R_) ? dbk[(size_t)(r0 + 1) * L_ + l] : 0.0f;
    acc = __builtin_amdgcn_wmma_f32_16x16x4_f32(false, a, false, bm,
                                                (short)0, acc, false, false);
  }

  const size_t bkD = (size_t)(b * K_ + k) * D_;
  #pragma unroll
  for (int v = 0; v < 8; ++v) {
    const int d = mt * 16 + v + 8 * half;
    const float dtv = acc[v] + dtb[k * D_ + d];
    // numerically stable softplus
    const float delta = (dtv > 0.0f) ? (dtv + log1pf(expf(-dtv)))
                                     : log1pf(expf(dtv));
    const float xv = xs[(bkD + d) * L_ + l];
    float2 pr; pr.x = delta; pr.y = delta * xv;
    *(float2*)(dd + ((bkD + d) * L_ + l) * 2) = pr;
  }
}

// ---------------------------------------------------------------------------
// Kernel 3: selective scan. Block = 8 waves, uniform (b,k); wave w handles
// channels d = 2*(dpg*8+w) + {0,1}; lanes 0..15 / 16..31 = 16 states of each.
// B/C stream (shared by all 8 waves) double-buffered via async-to-LDS.
// blocks = (B*K)*6 = 96
// ---------------------------------------------------------------------------
__global__ __launch_bounds__(256) void scan_kernel(const float* __restrict__ xs,
                                                   const float* __restrict__ dd,
                                                   const float* __restrict__ bc_t,
                                                   const float* __restrict__ A_logs,
                                                   const float* __restrict__ Ds,
                                                   float* __restrict__ y) {
  __shared__ float lbc[2][CHUNK * 32];

  const int bk   = blockIdx.x / 6;               // b*K + k
  const int dpg  = blockIdx.x % 6;
  const int wv   = threadIdx.x >> 5;
  const int lane = threadIdx.x & 31;
  const int half = lane >> 4;
  const int n    = lane & 15;

  const int dp = dpg * 8 + wv;
  const int d  = dp * 2 + half;
  const int k  = bk & 3;
  const int kd = k * D_ + d;

  const float Aval = -expf(A_logs[(size_t)kd * N_ + n]);   // A = -exp(A_logs)
  const float Dval = Ds[kd];

  const float* xrow  = xs   + ((size_t)bk * D_ + d) * L_;
  const float* ddrow = dd   + ((size_t)bk * D_ + d) * L_ * 2;
  const float* bcbk  = bc_t + (size_t)bk * L_ * 32;
  float*       yrow  = y    + ((size_t)bk * D_ + d) * L_;

  // --- async stage of one CHUNK of the B/C stream into LDS ------------------
  auto stage = [&](int c, int buf) {
    const int t = threadIdx.x;
    const unsigned lbase = (unsigned)(unsigned long long)(&lbc[buf][0]);
    const float* src = bcbk + (size_t)c * CHUNK * 32;
    #pragma unroll
    for (int i = 0; i < (CHUNK * 32 * 4) / (256 * 16); ++i) {   // 2 issues
      const float* g = src + (t + i * 256) * 4;                 // 16 B per lane
      const unsigned lo = lbase + (unsigned)((t + i * 256) * 16);
      asm volatile("global_load_async_to_lds_b128 %0, %1, off"
                   :: "v"(lo), "v"(g) : "memory");
    }
  };

  stage(0, 0);
  asm volatile("s_wait_asynccnt 0x0" ::: "memory");
  __syncthreads();

  float h = 0.0f;
  for (int c = 0; c < L_ / CHUNK; ++c) {
    const int buf = c & 1;
    if (c + 1 < L_ / CHUNK) stage(c + 1, buf ^ 1);

    const int l0 = c * CHUNK;
    #pragma unroll 4
    for (int j = 0; j < CHUNK; ++j) {
      const int l = l0 + j;
      const float2 dv = *(const float2*)(ddrow + 2 * l);   // {delta, delta*x}
      const float Bv = lbc[buf][j * 32 + n];
      const float Cv = lbc[buf][j * 32 + 16 + n];

      const float dA = __expf(dv.x * Aval);
      h = dA * h + dv.y * Bv;                  // inclusive scan, h[-1]=0

      float p = h * Cv;
      p += __shfl_xor(p, 1, 16);
      p += __shfl_xor(p, 2, 16);
      p += __shfl_xor(p, 4, 16);
      p += __shfl_xor(p, 8, 16);
      if (n == 0) yrow[l] = p + Dval * xrow[l];
    }

    asm volatile("s_wait_asynccnt 0x0" ::: "memory");
    __syncthreads();
  }
}

// ---------------------------------------------------------------------------
extern "C" void kernel_launch(void* const* d_in, const int* in_sizes, int n_in,
                              void* d_out, int out_size, void* d_ws, size_t ws_size,
                              hipStream_t stream) {
  const float* xs     = (const float*)d_in[0];  // (B,K,D,L)
  const float* A_logs = (const float*)d_in[1];  // (K*D, N)
  const float* Ds     = (const float*)d_in[2];  // (K*D)
  const float* dtw    = (const float*)d_in[3];  // (K,D,R)
  const float* dtb    = (const float*)d_in[4];  // (K,D)
  const float* xpw    = (const float*)d_in[5];  // (K,C,D)
  float* y  = (float*)d_out;                    // (B,K,D,L)
  float* ws = (float*)d_ws;

  float* dts_cl = ws;                           // B*K*8*L
  float* bc_t   = dts_cl + DTS_CL_ELEMS;        // B*K*L*32
  float* dd     = bc_t + BC_T_ELEMS;            // B*K*D*L*2

  proj1_kernel<<<192, 256, 0, stream>>>(xs, xpw, dts_cl, bc_t);
  proj2_kernel<<<1536, 256, 0, stream>>>(dts_cl, dtw, dtb, xs, dd);
  scan_kernel<<<96, 256, 0, stream>>>(xs, dd, bc_t, A_logs, Ds, y);
}